// GIN_87187836109073
// MI455X (gfx1250) — compile-verified
//
#include <hip/hip_runtime.h>

#define HID 64
#define MHID 128
#define ATOM_F 9
#define ATOM_V 119
#define EDGE_F 3
#define EDGE_V 22
#define OUT_COLS 256
#define BN_EPS 1e-5f

typedef float v2f __attribute__((ext_vector_type(2)));
typedef float v8f __attribute__((ext_vector_type(8)));

#if defined(__AMDGCN__) && __has_builtin(__builtin_amdgcn_wmma_f32_16x16x4_f32)
#define HAVE_WMMA_F32 1
#else
#define HAVE_WMMA_F32 0
#endif

__device__ __forceinline__ v8f wmma_16x16x4_f32(v2f a, v2f b, v8f c) {
#if HAVE_WMMA_F32
    // (neg_a, A, neg_b, B, c_mod, C, reuse_a, reuse_b)
    return __builtin_amdgcn_wmma_f32_16x16x4_f32(false, a, false, b, (short)0, c,
                                                 false, false);
#else
    (void)a; (void)b;
    return c;  // host pass / missing-builtin stub (dead on device)
#endif
}

// ---------------------------------------------------------------------------
// h[n,d] = sum_f atom_emb[f, x[n,f], d]; pooled0 += h
// ---------------------------------------------------------------------------
__global__ __launch_bounds__(256) void gin_embed_pool0_kernel(
    const int* __restrict__ x, const float* __restrict__ atom_emb,
    const int* __restrict__ batch, float* __restrict__ h,
    float* __restrict__ out, int n_nodes)
{
    long long t = (long long)blockIdx.x * 256 + threadIdx.x;
    int n = (int)(t >> 6);
    int d = (int)(t & 63);
    if (n >= n_nodes) return;
    float v = 0.0f;
#pragma unroll
    for (int f = 0; f < ATOM_F; ++f) {
        int idx = x[(long long)n * ATOM_F + f];
        v += atom_emb[((long long)f * ATOM_V + idx) * HID + d];
    }
    h[(long long)n * HID + d] = v;
    atomicAdd(&out[(long long)batch[n] * OUT_COLS + d], v);
}

// ---------------------------------------------------------------------------
// msg = relu(h[src,d] + sum_f edge_emb[f, attr[e,f], d]); agg[dst,d] += msg
// ---------------------------------------------------------------------------
__global__ __launch_bounds__(256) void gin_edge_kernel(
    const int* __restrict__ edge_index, const int* __restrict__ edge_attr,
    const float* __restrict__ h, const float* __restrict__ edge_emb_l,
    float* __restrict__ agg, int n_edges)
{
    long long t = (long long)blockIdx.x * 256 + threadIdx.x;
    long long e = t >> 6;
    int d = (int)(t & 63);
    if (e >= n_edges) return;
    int src = edge_index[e];
    int dst = edge_index[(long long)n_edges + e];
    int a0 = edge_attr[e * 3 + 0];
    int a1 = edge_attr[e * 3 + 1];
    int a2 = edge_attr[e * 3 + 2];
    float ev = edge_emb_l[(0 * EDGE_V + a0) * HID + d]
             + edge_emb_l[(1 * EDGE_V + a1) * HID + d]
             + edge_emb_l[(2 * EDGE_V + a2) * HID + d];
    float msg = fmaxf(h[(long long)src * HID + d] + ev, 0.0f);
    atomicAdd(&agg[(long long)dst * HID + d], msg);
}

// ---------------------------------------------------------------------------
// Per 16-node tile (one wave32 each):
//   z = relu((1+eps)*h + agg)                       [16x64]  -> LDS
//   y = relu(BN(z @ W1^T + b1))                     [16x128] -> LDS (WMMA)
//   h' = relu(y @ W2^T + b2)                        [16x64]  (WMMA)
//   h'-> global; pooled[layer+1] += h' (atomics)
// Fragment layouts per CDNA5 ISA 7.12.2:
//   A(16x4 f32): m=lane&15, VGPR{0,1} hold K={2*kh, 2*kh+1}, kh=lane>>4
//   B(4x16 f32): n=lane&15, same K mapping
//   C/D(16x16):  n=lane&15, VGPR r -> m = r + 8*kh
// ---------------------------------------------------------------------------
__global__ __launch_bounds__(128) void gin_mlp_pool_kernel(
    const float* __restrict__ hin, const float* __restrict__ agg,
    const float* __restrict__ lin1_w, const float* __restrict__ lin1_b,
    const float* __restrict__ bn_g, const float* __restrict__ bn_b,
    const float* __restrict__ bn_m, const float* __restrict__ bn_v,
    const float* __restrict__ lin2_w, const float* __restrict__ lin2_b,
    const float* __restrict__ eps, const int* __restrict__ batch,
    float* __restrict__ hout, float* __restrict__ out,
    int layer, int n_nodes)
{
    __shared__ float lds[4][16 * MHID];   // 4 waves x 8 KB = 32 KB

    const int wave = threadIdx.x >> 5;
    const int lane = threadIdx.x & 31;
    const int tiles = (n_nodes + 15) >> 4;
    int tile = blockIdx.x * 4 + wave;
    const bool active = tile < tiles;     // wave-uniform
    if (!active) tile = tiles - 1;        // clamp so reads stay in range
    const int node0 = tile << 4;

    const float epsv = 1.0f + eps[layer];
    float* __restrict__ Z = lds[wave];    // used as 16x64 (stride 64) for z

    // stage z into LDS
    for (int i = lane; i < 16 * HID; i += 32) {
        int m = i >> 6, d = i & 63;
        int node = node0 + m;
        if (node >= n_nodes) node = n_nodes - 1;
        long long off = (long long)node * HID + d;
        Z[m * HID + d] = fmaxf(fmaf(epsv, hin[off], agg[off]), 0.0f);
    }
    __syncthreads();

    const int mn = lane & 15;   // A row / B,C,D column within tile
    const int kh = lane >> 4;   // K-half select

    // ---- GEMM1: [16x64] x [64x128] via 8 col-tiles x 16 k-steps ----
    const float* __restrict__ W1 = lin1_w + (long long)layer * MHID * HID;
    v8f acc1[8];
    const v8f vzero = {0.f, 0.f, 0.f, 0.f, 0.f, 0.f, 0.f, 0.f};
#pragma unroll
    for (int jt = 0; jt < 8; ++jt) acc1[jt] = vzero;

    for (int kt = 0; kt < 16; ++kt) {
        const int k0 = kt * 4 + kh * 2;
        v2f a;
        a.x = Z[mn * HID + k0];
        a.y = Z[mn * HID + k0 + 1];
#pragma unroll
        for (int jt = 0; jt < 8; ++jt) {
            const int j = jt * 16 + mn;
            v2f b;
            b.x = W1[j * HID + k0];      // W1^T[k][j] = lin1_w[j][k]
            b.y = W1[j * HID + k0 + 1];
            acc1[jt] = wmma_16x16x4_f32(a, b, acc1[jt]);
        }
    }
    __syncthreads();   // all Z reads done before Y overwrites the region

    // ---- epilogue1: bias + BN(eval) + relu -> Y in LDS (stride 128) ----
    float* __restrict__ Y = lds[wave];
#pragma unroll
    for (int jt = 0; jt < 8; ++jt) {
        const int j = jt * 16 + mn;
        const float sc = bn_g[layer * MHID + j] * rsqrtf(bn_v[layer * MHID + j] + BN_EPS);
        const float sh = bn_b[layer * MHID + j] - bn_m[layer * MHID + j] * sc;
        const float bias = lin1_b[layer * MHID + j];
#pragma unroll
        for (int r = 0; r < 8; ++r) {
            const int m = r + 8 * kh;
            float v = acc1[jt][r] + bias;
            v = fmaxf(fmaf(v, sc, sh), 0.0f);
            Y[m * MHID + j] = v;
        }
    }
    __syncthreads();

    // ---- GEMM2: [16x128] x [128x64] via 4 col-tiles x 32 k-steps ----
    const float* __restrict__ W2 = lin2_w + (long long)layer * HID * MHID;
    v8f acc2[4];
#pragma unroll
    for (int jt = 0; jt < 4; ++jt) acc2[jt] = vzero;

    for (int kt = 0; kt < 32; ++kt) {
        const int k0 = kt * 4 + kh * 2;
        v2f a;
        a.x = Y[mn * MHID + k0];
        a.y = Y[mn * MHID + k0 + 1];
#pragma unroll
        for (int jt = 0; jt < 4; ++jt) {
            const int j = jt * 16 + mn;
            v2f b;
            b.x = W2[j * MHID + k0];     // W2^T[k][j] = lin2_w[j][k]
            b.y = W2[j * MHID + k0 + 1];
            acc2[jt] = wmma_16x16x4_f32(a, b, acc2[jt]);
        }
    }

    // ---- epilogue2: bias + relu -> h', pooled atomics ----
    const int colbase = HID * (layer + 1);
    if (active) {
#pragma unroll
        for (int r = 0; r < 8; ++r) {
            const int m = r + 8 * kh;
            const int node = node0 + m;
            if (node >= n_nodes) continue;
            const int g = batch[node];
#pragma unroll
            for (int jt = 0; jt < 4; ++jt) {
                const int j = jt * 16 + mn;
                float v = fmaxf(acc2[jt][r] + lin2_b[layer * HID + j], 0.0f);
                hout[(long long)node * HID + j] = v;
                atomicAdd(&out[(long long)g * OUT_COLS + colbase + j], v);
            }
        }
    }
}

// ---------------------------------------------------------------------------
extern "C" void kernel_launch(void* const* d_in, const int* in_sizes, int n_in,
                              void* d_out, int out_size, void* d_ws, size_t ws_size,
                              hipStream_t stream)
{
    const int*   x          = (const int*)d_in[0];
    const int*   edge_index = (const int*)d_in[1];
    const int*   edge_attr  = (const int*)d_in[2];
    const int*   batch      = (const int*)d_in[3];
    const float* atom_emb   = (const float*)d_in[4];
    const float* edge_emb   = (const float*)d_in[5];
    const float* lin1_w     = (const float*)d_in[6];
    const float* lin1_b     = (const float*)d_in[7];
    const float* bn_g       = (const float*)d_in[8];
    const float* bn_b       = (const float*)d_in[9];
    const float* bn_m       = (const float*)d_in[10];
    const float* bn_v       = (const float*)d_in[11];
    const float* lin2_w     = (const float*)d_in[12];
    const float* lin2_b     = (const float*)d_in[13];
    const float* eps        = (const float*)d_in[14];

    const int n_nodes = in_sizes[0] / ATOM_F;
    const int n_edges = in_sizes[2] / EDGE_F;

    float* h   = (float*)d_ws;
    float* agg = h + (size_t)n_nodes * HID;

    hipMemsetAsync(d_out, 0, (size_t)out_size * sizeof(float), stream);

    {
        long long total = (long long)n_nodes * HID;
        int blocks = (int)((total + 255) / 256);
        gin_embed_pool0_kernel<<<blocks, 256, 0, stream>>>(
            x, atom_emb, batch, h, (float*)d_out, n_nodes);
    }

    for (int l = 0; l < 3; ++l) {
        hipMemsetAsync(agg, 0, (size_t)n_nodes * HID * sizeof(float), stream);

        long long etotal = (long long)n_edges * HID;
        int eblocks = (int)((etotal + 255) / 256);
        gin_edge_kernel<<<eblocks, 256, 0, stream>>>(
            edge_index, edge_attr, h,
            edge_emb + (size_t)l * EDGE_F * EDGE_V * HID, agg, n_edges);

        int tiles = (n_nodes + 15) / 16;
        int mblocks = (tiles + 3) / 4;
        gin_mlp_pool_kernel<<<mblocks, 128, 0, stream>>>(
            h, agg, lin1_w, lin1_b, bn_g, bn_b, bn_m, bn_v,
            lin2_w, lin2_b, eps, batch, h, (float*)d_out, l, n_nodes);
    }
}